// H_02_Linear_cla_heterogeneous_batch_87866440941679
// MI455X (gfx1250) — compile-verified
//
#include <hip/hip_runtime.h>
#include <hip/hip_bf16.h>

// ---------------------------------------------------------------------------
// logits[i] = x[i] @ W[sid[i]].T + b[sid[i]]
// B=262144, D=256, S=16, C=10, fp32.
//
// Memory-bound (≈266MB @ 23.3 TB/s ≈ 11.4us). Pipeline:
//   1) bin sample indices by system (LDS-aggregated histogram + scatter)
//   2) one wave32 per 16-sample tile of one system:
//        - TDM gather-mode tensor_load_to_lds pulls the 16 gathered 1KB
//          x-rows into LDS (32-bit row indices, 8 rows per D#, 2 D# per
//          tile), D# pad (+4 dwords / 256 dwords) -> 260-dword row stride
//          so the wave's ds_load_b64 A-fragment reads hit all 64 banks.
//        - 64x V_WMMA_F32_16X16X4_F32 accumulate over K=256 (exact FLOPs).
//          A-fragment reads are explicit ds_load_b64 (manual s_wait_dscnt),
//          issued 16-at-a-time per 16-WMMA chunk. All tile control flow is
//          scalar (readfirstlane) so EXEC is all-1s at every WMMA.
// ---------------------------------------------------------------------------

typedef __attribute__((ext_vector_type(2))) float        v2f;
typedef __attribute__((ext_vector_type(8))) float        v8f;
typedef __attribute__((ext_vector_type(4))) unsigned int v4u;
typedef __attribute__((ext_vector_type(4))) int          v4i;
typedef __attribute__((ext_vector_type(8))) int          v8i;

#define DK   256
#define SN   16
#define CN   10

#define M_COUNT   0
#define M_BSTART  16
#define M_TOFF    33
#define M_FILL    50
#define META_INTS 128

// padded row stride in LDS (floats): 256 data + 4 pad dwords -> 64-bank spread
#define ROWF   260
#define WAVEF  (16 * ROWF)          // floats per wave tile region
#define WAVES_PER_BLOCK 2

__global__ void k_zero_meta(int* __restrict__ meta) {
    if (threadIdx.x < META_INTS) meta[threadIdx.x] = 0;
}

__global__ void k_histo(const int* __restrict__ sid, int* __restrict__ meta, int Bn) {
    __shared__ int lh[SN];
    if (threadIdx.x < SN) lh[threadIdx.x] = 0;
    __syncthreads();
    int stride = gridDim.x * blockDim.x;
    for (int i = blockIdx.x * blockDim.x + threadIdx.x; i < Bn; i += stride)
        atomicAdd(&lh[sid[i]], 1);
    __syncthreads();
    if (threadIdx.x < SN) atomicAdd(&meta[M_COUNT + threadIdx.x], lh[threadIdx.x]);
}

__global__ void k_scan(int* __restrict__ meta) {
    if (threadIdx.x == 0) {
        int bs = 0, to = 0;
        for (int s = 0; s < SN; ++s) {
            meta[M_BSTART + s] = bs;
            meta[M_TOFF + s]   = to;
            int c = meta[M_COUNT + s];
            bs += c;
            to += (c + 15) >> 4;
        }
        meta[M_BSTART + SN] = bs;
        meta[M_TOFF + SN]   = to;   // total tiles
    }
}

__global__ void k_scatter(const int* __restrict__ sid, int* __restrict__ meta,
                          int* __restrict__ idxArr, int Bn) {
    __shared__ int lh[SN];
    __shared__ int lbase[SN];
    if (threadIdx.x < SN) lh[threadIdx.x] = 0;
    __syncthreads();
    int stride = gridDim.x * blockDim.x;
    int myPos[8], myS[8], myI[8];
    int cnt = 0;
    for (int i = blockIdx.x * blockDim.x + threadIdx.x; i < Bn && cnt < 8; i += stride) {
        int s = sid[i];
        myS[cnt] = s; myI[cnt] = i;
        myPos[cnt] = atomicAdd(&lh[s], 1);
        ++cnt;
    }
    __syncthreads();
    if (threadIdx.x < SN)
        lbase[threadIdx.x] = atomicAdd(&meta[M_FILL + threadIdx.x], lh[threadIdx.x]);
    __syncthreads();
    for (int e = 0; e < cnt; ++e) {
        int s = myS[e];
        idxArr[meta[M_BSTART + s] + lbase[s] + myPos[e]] = myI[e];
    }
}

// Issue one gather-mode TDM load: 8 rows of 256 f32 from x into LDS.
// ldsByteOff is an offset within the workgroup LDS allocation.
static __device__ __forceinline__
void tdm_gather8(const float* xbase, unsigned ldsByteOff, int Bn,
                 int i0, int i1, int i2, int i3, int i4, int i5, int i6, int i7) {
    unsigned long long ga = (unsigned long long)(uintptr_t)xbase;
    // group0: count=1, gather_index_size=1 (32b), gather_mode=1, lds_addr,
    //         global_addr (57b), type=2
    v4u g0 = { 0xC0000001u,
               ldsByteOff,
               (unsigned)(ga & 0xFFFFFFFFu),
               (unsigned)(((ga >> 32) & 0x1FFFFFFu) | (2u << 30)) };
    // group1: data_size=4B, pad_enable, pad_interval=7 (256 dwords),
    //         pad_amount=3 (4 dwords); tensor_dim0=256, tensor_dim1=Bn,
    //         tile_dim0=256, tile_dim1=8 valid indices, dim0_stride=256
    v8i g1 = { (int)((2u << 16) | (1u << 20) | (7u << 22) | (3u << 25)),
               (int)(256u << 16),
               (int)(((unsigned)Bn & 0xFFFFu) << 16),
               (int)((((unsigned)Bn >> 16) & 0xFFFFu) | (256u << 16)),
               8, 256, 0, 0 };
    v4i g2 = { i0, i1, i2, i3 };
    v4i g3 = { i4, i5, i6, i7 };
#if __clang_major__ >= 23
    v8i g4 = {};
    __builtin_amdgcn_tensor_load_to_lds(g0, g1, g2, g3, g4, 0);
#else
    __builtin_amdgcn_tensor_load_to_lds(g0, g1, g2, g3, 0);
#endif
}

__global__ __launch_bounds__(32 * WAVES_PER_BLOCK)
void k_gemv_wmma(const float* __restrict__ x, const float* __restrict__ W,
                 const float* __restrict__ bias, const int* __restrict__ meta,
                 const int* __restrict__ idxArr, float* __restrict__ out, int Bn) {
    __shared__ float sh[WAVES_PER_BLOCK * WAVEF];   // 33,280 B

    const int wave = threadIdx.x >> 5;
    const int lane = threadIdx.x & 31;
    // wave-uniform tile id forced into an SGPR -> scalar control flow
    const int tileId = __builtin_amdgcn_readfirstlane(
        (int)(blockIdx.x * WAVES_PER_BLOCK + wave));

    const int total = meta[M_TOFF + SN];
    if (tileId >= total) return;                 // scalar, wave-uniform exit

    int s = 0;
    while (s < SN - 1 && tileId >= meta[M_TOFF + s + 1]) ++s;
    const int t         = tileId - meta[M_TOFF + s];
    const int cnt       = meta[M_COUNT + s];
    int r               = cnt - (t << 4);
    if (r > 16) r = 16;
    const int tileStart = meta[M_BSTART + s] + (t << 4);

    const int mA   = lane & 15;
    const int half = lane >> 4;
    const int ko   = half << 1;

    // B-fragment column (branch-free N-padding via 0/1 mask)
    const int   n     = mA;
    const float wmask = (n < CN) ? 1.0f : 0.0f;
    const int   nCl   = (n < CN) ? n : 0;
    const float* wrow = W + (size_t)(s * CN + nCl) * DK;

    // LDS byte offset of our shared array: low 32 bits of the shared-aperture
    // flat address are the LDS offset (ISA 10.2); also keeps sh referenced.
    const unsigned shBase  = (unsigned)(uintptr_t)(void*)sh;
    const unsigned ldsWave = shBase + (unsigned)(wave * WAVEF) * 4u;

    // ---- stage 16 gathered rows into LDS via TDM gather mode ----
    int ri[16];
#pragma unroll
    for (int m = 0; m < 16; ++m) {
        int mC = (m < r) ? m : (r - 1);          // clamp tail to a valid row
        ri[m] = __builtin_amdgcn_readfirstlane(idxArr[tileStart + mC]);
    }
    tdm_gather8(x, ldsWave,                  Bn, ri[0], ri[1], ri[2],  ri[3],
                                                ri[4], ri[5], ri[6],  ri[7]);
    tdm_gather8(x, ldsWave + 8u * ROWF * 4u, Bn, ri[8], ri[9], ri[10], ri[11],
                                                ri[12], ri[13], ri[14], ri[15]);
    __builtin_amdgcn_s_wait_tensorcnt((short)0);
    asm volatile("" ::: "memory");               // keep LDS reads after the wait

    // per-lane A-fragment base: row mA, K phase ko (bank-conflict-free: 260f row)
    const unsigned ldsA = ldsWave + (unsigned)(mA * ROWF + ko) * 4u;

    v8f acc = {};
#pragma unroll
    for (int kc = 0; kc < DK; kc += 64) {        // 4 chunks of 16 k-steps
        v2f frag[16];
#pragma unroll
        for (int u = 0; u < 16; ++u)
            asm volatile("ds_load_b64 %0, %1 offset:%2"
                         : "=v"(frag[u])
                         : "v"(ldsA + (unsigned)kc * 4u), "i"(u * 16));
        asm volatile("s_wait_dscnt 0x0" ::: "memory");
#pragma unroll
        for (int u = 0; u < 16; ++u) {
            const int k = kc + 4 * u;
            v2f wv = *(const v2f*)(wrow + k + ko);   // L2-resident W
            wv *= wmask;
            acc = __builtin_amdgcn_wmma_f32_16x16x4_f32(
                false, frag[u], false, wv, (short)0, acc, false, false);
        }
    }

    // D layout: VGPR j, lanes 0-15 -> row j, lanes 16-31 -> row j+8, col=lane&15
    if (n < CN) {
        const float bn = bias[s * CN + n];
#pragma unroll
        for (int j = 0; j < 8; ++j) {
            int m = j + (half << 3);
            if (m < r) {
                int row = idxArr[tileStart + m];
                out[(size_t)row * CN + n] = acc[j] + bn;
            }
        }
    }
}

// safety net if workspace is too small for the index array
__global__ void k_gemv_fallback(const float* __restrict__ x, const int* __restrict__ sid,
                                const float* __restrict__ W, const float* __restrict__ bias,
                                float* __restrict__ out, int Bn) {
    int i = blockIdx.x * blockDim.x + threadIdx.x;
    if (i >= Bn) return;
    int s = sid[i];
    const float* xr = x + (size_t)i * DK;
    const float* Ws = W + (size_t)s * CN * DK;
    float acc[CN];
#pragma unroll
    for (int c = 0; c < CN; ++c) acc[c] = bias[s * CN + c];
    for (int k = 0; k < DK; k += 4) {
        float4 xv = *(const float4*)(xr + k);
#pragma unroll
        for (int c = 0; c < CN; ++c) {
            float4 wv = *(const float4*)(Ws + (size_t)c * DK + k);
            acc[c] += xv.x * wv.x + xv.y * wv.y + xv.z * wv.z + xv.w * wv.w;
        }
    }
#pragma unroll
    for (int c = 0; c < CN; ++c) out[(size_t)i * CN + c] = acc[c];
}

extern "C" void kernel_launch(void* const* d_in, const int* in_sizes, int n_in,
                              void* d_out, int out_size, void* d_ws, size_t ws_size,
                              hipStream_t stream) {
    const float* x    = (const float*)d_in[0];
    const int*   sid  = (const int*)d_in[1];
    const float* W    = (const float*)d_in[2];
    const float* bias = (const float*)d_in[3];
    float*       out  = (float*)d_out;
    const int    Bn   = in_sizes[1];

    const size_t need = (size_t)(META_INTS + Bn) * sizeof(int);
    if (ws_size < need) {
        k_gemv_fallback<<<(Bn + 255) / 256, 256, 0, stream>>>(x, sid, W, bias, out, Bn);
        return;
    }

    int* meta   = (int*)d_ws;
    int* idxArr = meta + META_INTS;

    k_zero_meta<<<1, 128, 0, stream>>>(meta);
    k_histo<<<256, 256, 0, stream>>>(sid, meta, Bn);
    k_scan<<<1, 32, 0, stream>>>(meta);
    {
        int blocks = (Bn + 256 * 4 - 1) / (256 * 4);   // per-thread cap of 8
        k_scatter<<<blocks, 256, 0, stream>>>(sid, meta, idxArr, Bn);
    }
    {
        const int maxTiles = (Bn >> 4) + SN;
        int blocks = (maxTiles + WAVES_PER_BLOCK - 1) / WAVES_PER_BLOCK;
        k_gemv_wmma<<<blocks, 32 * WAVES_PER_BLOCK, 0, stream>>>(
            x, W, bias, meta, idxArr, out, Bn);
    }
}